// costvol_layer_23802708755022
// MI455X (gfx1250) — compile-verified
//
#include <hip/hip_runtime.h>

typedef __attribute__((ext_vector_type(2))) float v2f;
typedef __attribute__((ext_vector_type(8))) float v8f;

#define SEARCH_RANGE 4
#define MAXOFF 9                  // 2r+1
#define NOFF   81                 // MAXOFF^2
#define C_TOT  128
#define H_TOT  128
#define W_TOT  128
#define KCHUNK 32                 // channels staged per pass
#define NKC    4                  // C_TOT / KCHUNK
#define WTILE  16
#define NCOLS  24                 // WTILE + 2*SEARCH_RANGE (96B rows: b128-aligned)
#define NTHREADS 288              // 9 wave32, wave y owns y-offset y

// Wait until this wave's ASYNCcnt <= n (retires older async-to-LDS batches).
#define S_WAIT_ASYNC(n) asm volatile("s_wait_asynccnt " #n ::: "memory")

// Low 32 bits of a generic LDS pointer == byte offset within the WG's LDS.
__device__ __forceinline__ unsigned lds_addr32(const void* p) {
  return (unsigned)(unsigned long long)(uintptr_t)p;
}

// GLOBAL_LOAD_ASYNC_TO_LDS_B128, GVS mode: SGPR 64b base + 32b per-lane offset.
// LDS dest byte address per lane in a VGPR; tracked by ASYNCcnt.
__device__ __forceinline__ void async_b128(unsigned lds_dst, unsigned voff,
                                           unsigned long long sbase) {
  asm volatile("global_load_async_to_lds_b128 %0, %1, %2"
               :: "v"(lds_dst), "v"(voff), "s"(sbase)
               : "memory");
}

// Stage one K-chunk. Wave wv loads B row ry=wv (lane = channel, 6 column
// quads); waves 0..3 additionally load one A quad each. Per-wave ASYNCcnt
// per chunk: 7 (wv<4) or 6 (wv>=4) -- uniform, so waits can be exact.
// Zero-padding: OOB quads keep their LDS destination but read a zero page.
__device__ __forceinline__ void issue_chunk(unsigned long long firstb,
                                            unsigned long long secondb,
                                            unsigned long long zbase,
                                            float* AsBuf, float* BsBuf,
                                            int h, int w0, int kc,
                                            int wv, int lane) {
  // ---- B: second[b, kc*32+lane, h+wv-4, w0-4 + 4*n4 ..+3] ----
  const int gr = h + wv - SEARCH_RANGE;                   // wave-uniform
  const bool rowOK = (unsigned)gr < (unsigned)H_TOT;
  const unsigned bvoff0 =
      (unsigned)((((kc * KCHUNK + lane) * H_TOT + gr) * W_TOT) * 4);
  const unsigned bdst0 =
      lds_addr32(BsBuf + (wv * KCHUNK + lane) * NCOLS);
  const unsigned zoff = (unsigned)(lane << 4);            // 16B/lane of zeros
#pragma unroll
  for (int n4 = 0; n4 < NCOLS / 4; ++n4) {
    const int gc = w0 + n4 * 4 - SEARCH_RANGE;            // instruction-uniform
    const bool ok = rowOK && ((unsigned)gc < (unsigned)W_TOT);
    unsigned long long base = ok ? secondb : zbase;       // scalar select
    unsigned voff = ok ? (bvoff0 + (unsigned)(gc * 4)) : zoff;
    async_b128(bdst0 + (unsigned)(n4 * 16), voff, base);
  }
  // ---- A: first[b, kc*32+ak, h, w0 + 4*am4 ..+3] (waves 0..3, always valid) ----
  if (wv < 4) {
    const int q   = wv * 32 + lane;                       // 0..127
    const int ak  = q >> 2;
    const int am4 = q & 3;
    const unsigned dstA  = lds_addr32(AsBuf + ak * WTILE + am4 * 4);
    const unsigned voffA =
        (unsigned)(((((kc * KCHUNK + ak) * H_TOT + h) * W_TOT) + w0 + am4 * 4) * 4);
    async_b128(dstA, voffA, firstb);
  }
}

__global__ __launch_bounds__(NTHREADS)
void costvol_wmma_async_kernel(const float* __restrict__ first,
                               const float* __restrict__ second,
                               const float* __restrict__ zbuf,
                               float* __restrict__ out) {
  // Async-friendly layouts (LDS byte order == global byte order per quad):
  //   As[buf][k][m]    : 32 x 16 f32
  //   Bs[buf][ry][k][n]: 9 x 32 x 24 f32
  __shared__ __align__(16) float As[2][KCHUNK * WTILE];           //  4 KB
  __shared__ __align__(16) float Bs[2][MAXOFF * KCHUNK * NCOLS];  // 54 KB
  __shared__ __align__(16) float Ds[MAXOFF * WTILE * NCOLS];      // 13.5 KB

  const int wtile = blockIdx.x;   // 0..7
  const int h     = blockIdx.y;   // 0..127
  const int b     = blockIdx.z;   // 0..7
  const int w0    = wtile * WTILE;

  const int tid  = threadIdx.x;
  const int lane = tid & 31;
  const int hi   = lane >> 4;
  const int lo   = lane & 15;
  // Scalarized wave id (0..8): lane 0's tid >> 5.
  const int wv   = __builtin_amdgcn_readfirstlane(tid) >> 5;

  const size_t in_b = (size_t)b * C_TOT * H_TOT * W_TOT;
  const unsigned long long firstb  = (unsigned long long)(uintptr_t)(first + in_b);
  const unsigned long long secondb = (unsigned long long)(uintptr_t)(second + in_b);
  const unsigned long long zbase   = (unsigned long long)(uintptr_t)zbuf;

  v8f acc0 = {};  // D columns 0..15
  v8f acc1 = {};  // D columns 16..23 (upper half unused)

  // ---- software pipeline: prime chunk 0, overlap load(kc+1) with compute(kc)
  issue_chunk(firstb, secondb, zbase, As[0], Bs[0], h, w0, /*kc=*/0, wv, lane);

  int buf = 0;
  for (int kc = 0; kc < NKC; ++kc) {
    if (kc + 1 < NKC) {
      issue_chunk(firstb, secondb, zbase, As[buf ^ 1], Bs[buf ^ 1],
                  h, w0, kc + 1, wv, lane);
      if (wv < 4) { S_WAIT_ASYNC(7); } else { S_WAIT_ASYNC(6); }
    } else {
      S_WAIT_ASYNC(0);
    }
    __syncthreads();              // all waves' chunk-kc data visible in LDS

    // ---- WMMA: wave wv accumulates its y-offset over this K chunk ----
    const float* Asb = As[buf];
    const float* Bsb = Bs[buf] + wv * (KCHUNK * NCOLS);
#pragma unroll
    for (int kk2 = 0; kk2 < KCHUNK / 4; ++kk2) {   // 8 steps of K=4
      int k = kk2 * 4 + 2 * hi;                    // half-wave K split
      v2f a, b0, b1;
      a.x  = Asb[k * WTILE + lo];                  // A[m=lo, K=k]
      a.y  = Asb[(k + 1) * WTILE + lo];            // A[m=lo, K=k+1]
      b0.x = Bsb[k * NCOLS + lo];                  // B[K=k,   n=lo]
      b0.y = Bsb[(k + 1) * NCOLS + lo];
      b1.x = Bsb[k * NCOLS + 16 + lo];             // B[K=k,   n=16+lo]
      b1.y = Bsb[(k + 1) * NCOLS + 16 + lo];
      acc0 = __builtin_amdgcn_wmma_f32_16x16x4_f32(false, a, false, b0,
                                                   (short)0, acc0, false, false);
      acc1 = __builtin_amdgcn_wmma_f32_16x16x4_f32(false, a, false, b1,
                                                   (short)0, acc1, false, false);
    }
    __syncthreads();              // done reading buf before it is overwritten
    buf ^= 1;
  }

  // ---- re-layout D through LDS ----
  // C/D layout: lane L, VGPR i -> D[M = i + 8*(L>>4), N = (L&15) + 16*tile]
#pragma unroll
  for (int i = 0; i < 8; ++i) {
    int M = i + 8 * hi;
    Ds[(wv * WTILE + M) * NCOLS + lo] = acc0[i];
    if (lo < NCOLS - 16)
      Ds[(wv * WTILE + M) * NCOLS + 16 + lo] = acc1[i];
  }
  __syncthreads();

  // ---- coalesced store: out[b, y*9+x, h, w0+m] = D[y][m][m+x] / 128 ----
  const float scale = 1.0f / (float)C_TOT;
  for (int idx = tid; idx < NOFF * WTILE; idx += NTHREADS) {
    int m = idx & 15;
    int o = idx >> 4;             // 0..80
    int y = o / MAXOFF;
    int x = o - y * MAXOFF;
    float v = Ds[(y * WTILE + m) * NCOLS + (m + x)] * scale;
    out[(((size_t)b * NOFF + o) * H_TOT + h) * W_TOT + (w0 + m)] = v;
  }
}

extern "C" void kernel_launch(void* const* d_in, const int* in_sizes, int n_in,
                              void* d_out, int out_size, void* d_ws, size_t ws_size,
                              hipStream_t stream) {
  (void)in_sizes; (void)n_in; (void)out_size;
  const float* first  = (const float*)d_in[0];
  const float* second = (const float*)d_in[1];
  float* out = (float*)d_out;

  // 512B zero page in workspace: read target for zero-pad async loads.
  size_t zb = ws_size < 512 ? ws_size : 512;
  if (zb) hipMemsetAsync(d_ws, 0, zb, stream);

  dim3 grid(W_TOT / WTILE, H_TOT, 8 /*B*/);
  dim3 block(NTHREADS);
  costvol_wmma_async_kernel<<<grid, block, 0, stream>>>(
      first, second, (const float*)d_ws, out);
}